// ScatterCFGEncodedNGramsToCFGNodeEncodings_32134945308872
// MI455X (gfx1250) — compile-verified
//
#include <hip/hip_runtime.h>
#include <math.h>

#define D 128
#define EPSV 1e-9f
#define LDS_STRIDE 130   // 128 + 2: row r, col c -> bank (2r + c) % 64, conflict-free

typedef __attribute__((ext_vector_type(2))) float v2f;
typedef __attribute__((ext_vector_type(4))) float v4f;
typedef __attribute__((ext_vector_type(8))) float v8f;

// IEEE-ordered float atomic max via int/uint monotone punning (valid for mixed
// signs when the location is initialized to -inf).
__device__ __forceinline__ void atomic_max_f32(float* addr, float v) {
    if (v >= 0.0f) {
        atomicMax((int*)addr, __float_as_int(v));
    } else {
        atomicMin((unsigned int*)addr, __float_as_uint(v));
    }
}

// ---------------------------------------------------------------------------
// 0) init: agg (in d_out) = 0, seg_max = -inf, denom = 0
// ---------------------------------------------------------------------------
__global__ void init_kernel(float* __restrict__ agg, float* __restrict__ seg_max,
                            float* __restrict__ denom, int N) {
    size_t t = (size_t)blockIdx.x * blockDim.x + threadIdx.x;
    size_t tot = (size_t)N * D;
    if (t < tot) agg[t] = 0.0f;
    if (t < (size_t)N) {
        seg_max[t] = -__builtin_inff();
        denom[t]   = 0.0f;
    }
}

// ---------------------------------------------------------------------------
// 1) keys = prev @ attn_W + attn_b   (M=N_nodes, K=128, N=128)
// Block = 16-row stripe; 8 waves each own a 16x16 C tile (col tile = wave id).
// A tile (16x128) staged in LDS via coalesced float4 loads; unguarded loads are
// safe because OOB A-rows only affect OOB D-rows, whose stores are guarded.
// WMMA f32 16x16x4 fragment layouts per ISA:
//   A: lanes 0-15 -> M, v0=K0/v1=K1; lanes 16-31 -> v0=K2/v1=K3
//   B: lanes hold N, same K split.  C/D: v[r] = C[r + 8*half][n]
// ---------------------------------------------------------------------------
__global__ void __launch_bounds__(256) keys_kernel(
        const float* __restrict__ prev, const float* __restrict__ W,
        const float* __restrict__ bias, float* __restrict__ keys, int N) {
    __shared__ float As[16 * LDS_STRIDE];

    const int tid  = threadIdx.x;
    const int wave = tid >> 5;
    const int lane = tid & 31;
    const int half = lane >> 4;
    const int m    = lane & 15;
    const int row0 = blockIdx.x << 4;
    const int col0 = wave << 4;
    const int koff = half * 2;

    // Cooperative stage of the 16x128 A tile (2048 floats, 8 per thread).
#pragma unroll
    for (int i = 0; i < 2; ++i) {
        int pos = (tid + i * 256) * 4;            // 0..2044, step 4
        int r = pos >> 7;                          // / 128
        int c = pos & 127;
        int row = row0 + r;
        const float* src = prev + (size_t)(row < N ? row : 0) * D + c;
        v4f v = *(const v4f*)src;
        As[r * LDS_STRIDE + c + 0] = v.x;
        As[r * LDS_STRIDE + c + 1] = v.y;
        As[r * LDS_STRIDE + c + 2] = v.z;
        As[r * LDS_STRIDE + c + 3] = v.w;
    }
    __syncthreads();

    const float* Wc = W + col0 + m;                // column col0+m of W
    const float* Ar = As + m * LDS_STRIDE + koff;  // row m, K-half offset

    v8f acc = {};
#pragma unroll 8
    for (int k = 0; k < D; k += 4) {
        v2f a, b;
        a.x = Ar[k];
        a.y = Ar[k + 1];
        b.x = Wc[(size_t)(k + koff) * D];
        b.y = Wc[(size_t)(k + koff + 1) * D];
        acc = __builtin_amdgcn_wmma_f32_16x16x4_f32(false, a, false, b,
                                                    (short)0, acc, false, false);
    }

    const int n = col0 + m;
    const float bn = bias[n];
    if (row0 + 16 <= N) {                          // uniform fast path: full tile
#pragma unroll
        for (int r = 0; r < 8; ++r) {
            int row = row0 + r + half * 8;
            keys[(size_t)row * D + n] = acc[r] + bn;
        }
    } else {                                       // ragged last tile only
#pragma unroll
        for (int r = 0; r < 8; ++r) {
            int row = row0 + r + half * 8;
            if (row < N) keys[(size_t)row * D + n] = acc[r] + bn;
        }
    }
}

// ---------------------------------------------------------------------------
// 2) scores[e] = dot(vals[e], keys[idx[e]]) / sqrt(D); seg_max = atomic max.
// One wave per edge; lane L loads float4 at column 4L -> 512B coalesced row.
// Edge values are streamed non-temporally (read-once); keys stay L2-resident.
// ---------------------------------------------------------------------------
__global__ void __launch_bounds__(256) scores_kernel(
        const float* __restrict__ occ3, const float* __restrict__ occ5,
        const int* __restrict__ idx3, const int* __restrict__ idx5,
        const float* __restrict__ keys, float* __restrict__ scores,
        float* __restrict__ seg_max, long long E3, long long E) {
    long long e = (long long)blockIdx.x * 8 + (threadIdx.x >> 5);
    const int lane = threadIdx.x & 31;
    if (e >= E) return;

    const float* vrow;
    int idx;
    if (e < E3) { vrow = occ3 + (size_t)e * D;  idx = idx3[e]; }
    else        { long long e5 = e - E3;
                  vrow = occ5 + (size_t)e5 * D; idx = idx5[e5]; }

    const v4f v = __builtin_nontemporal_load((const v4f*)vrow + lane);
    const v4f k = *((const v4f*)(keys + (size_t)idx * D) + lane);
    float s = v.x * k.x + v.y * k.y + v.z * k.z + v.w * k.w;
    s += __shfl_xor(s, 16);
    s += __shfl_xor(s, 8);
    s += __shfl_xor(s, 4);
    s += __shfl_xor(s, 2);
    s += __shfl_xor(s, 1);
    if (lane == 0) {
        s *= 0.0883883476483184405f;   // 1/sqrt(128)
        scores[e] = s;
        atomic_max_f32(&seg_max[idx], s);
    }
}

// ---------------------------------------------------------------------------
// 3) ex[e] = exp(scores[e] - seg_max[idx]); denom[idx] += ex  (in-place)
// ---------------------------------------------------------------------------
__global__ void __launch_bounds__(256) ex_kernel(
        const int* __restrict__ idx3, const int* __restrict__ idx5,
        float* __restrict__ scores, const float* __restrict__ seg_max,
        float* __restrict__ denom, long long E3, long long E) {
    long long e = (long long)blockIdx.x * blockDim.x + threadIdx.x;
    if (e >= E) return;
    int idx = (e < E3) ? idx3[e] : idx5[e - E3];
    float ex = expf(scores[e] - seg_max[idx]);
    scores[e] = ex;
    atomicAdd(&denom[idx], ex);
}

// ---------------------------------------------------------------------------
// 4) agg[idx] += (ex/(denom+eps)) * vals   — wave per edge, float4 per lane.
// ---------------------------------------------------------------------------
__global__ void __launch_bounds__(256) agg_kernel(
        const float* __restrict__ occ3, const float* __restrict__ occ5,
        const int* __restrict__ idx3, const int* __restrict__ idx5,
        const float* __restrict__ ex, const float* __restrict__ denom,
        float* __restrict__ agg, long long E3, long long E) {
    long long e = (long long)blockIdx.x * 8 + (threadIdx.x >> 5);
    const int lane = threadIdx.x & 31;
    if (e >= E) return;

    const float* vrow;
    int idx;
    if (e < E3) { vrow = occ3 + (size_t)e * D;  idx = idx3[e]; }
    else        { long long e5 = e - E3;
                  vrow = occ5 + (size_t)e5 * D; idx = idx5[e5]; }

    const float p = ex[e] / (denom[idx] + EPSV);
    const v4f v = __builtin_nontemporal_load((const v4f*)vrow + lane);
    float* arow = agg + (size_t)idx * D + lane * 4;
    atomicAdd(arow + 0, p * v.x);
    atomicAdd(arow + 1, p * v.y);
    atomicAdd(arow + 2, p * v.z);
    atomicAdd(arow + 3, p * v.w);
}

// ---------------------------------------------------------------------------
// 5) z = sigmoid([prev;agg] @ gate_W + gate_b); out = z*agg + (1-z)*prev.
// agg lives in d_out. Both 16x128 A tiles (prev, agg) are staged to LDS before
// the single __syncthreads(); after it there are NO further global reads of
// agg, so overwriting d_out is hazard-free. Epilogue reads prev/agg from LDS.
// ---------------------------------------------------------------------------
__global__ void __launch_bounds__(256) gate_kernel(
        const float* __restrict__ prev, const float* __restrict__ gW,
        const float* __restrict__ gb, float* __restrict__ out, int N) {
    __shared__ float As[16 * LDS_STRIDE];   // prev tile
    __shared__ float Aa[16 * LDS_STRIDE];   // agg tile

    const int tid  = threadIdx.x;
    const int wave = tid >> 5;
    const int lane = tid & 31;
    const int half = lane >> 4;
    const int m    = lane & 15;
    const int row0 = blockIdx.x << 4;
    const int col0 = wave << 4;
    const int koff = half * 2;

#pragma unroll
    for (int i = 0; i < 2; ++i) {
        int pos = (tid + i * 256) * 4;
        int r = pos >> 7;
        int c = pos & 127;
        int row = row0 + r;
        size_t srow = (size_t)(row < N ? row : 0) * D + c;
        v4f vp = *(const v4f*)(prev + srow);
        v4f va = *(const v4f*)(out + srow);
        As[r * LDS_STRIDE + c + 0] = vp.x;
        As[r * LDS_STRIDE + c + 1] = vp.y;
        As[r * LDS_STRIDE + c + 2] = vp.z;
        As[r * LDS_STRIDE + c + 3] = vp.w;
        Aa[r * LDS_STRIDE + c + 0] = va.x;
        Aa[r * LDS_STRIDE + c + 1] = va.y;
        Aa[r * LDS_STRIDE + c + 2] = va.z;
        Aa[r * LDS_STRIDE + c + 3] = va.w;
    }
    __syncthreads();   // after this point: no global reads of agg (d_out)

    const float* Wc = gW + col0 + m;
    const float* Pr = As + m * LDS_STRIDE + koff;
    const float* Ag = Aa + m * LDS_STRIDE + koff;

    v8f acc = {};
#pragma unroll 8
    for (int k = 0; k < D; k += 4) {            // K = 0..127 : prev
        v2f a, b;
        a.x = Pr[k];
        a.y = Pr[k + 1];
        b.x = Wc[(size_t)(k + koff) * D];
        b.y = Wc[(size_t)(k + koff + 1) * D];
        acc = __builtin_amdgcn_wmma_f32_16x16x4_f32(false, a, false, b,
                                                    (short)0, acc, false, false);
    }
#pragma unroll 8
    for (int k = 0; k < D; k += 4) {            // K = 128..255 : agg
        v2f a, b;
        a.x = Ag[k];
        a.y = Ag[k + 1];
        b.x = Wc[(size_t)(D + k + koff) * D];
        b.y = Wc[(size_t)(D + k + koff + 1) * D];
        acc = __builtin_amdgcn_wmma_f32_16x16x4_f32(false, a, false, b,
                                                    (short)0, acc, false, false);
    }

    const int n = col0 + m;
    const float bn = gb[n];
    if (row0 + 16 <= N) {                        // uniform fast path: full tile
#pragma unroll
        for (int r = 0; r < 8; ++r) {
            int rl  = r + half * 8;              // local row
            int row = row0 + rl;
            float x = acc[r] + bn;
            float z = 1.0f / (1.0f + expf(-x));
            float a = Aa[rl * LDS_STRIDE + n];   // agg from LDS
            float p = As[rl * LDS_STRIDE + n];   // prev from LDS
            out[(size_t)row * D + n] = z * a + (1.0f - z) * p;
        }
    } else {                                     // ragged last tile only
#pragma unroll
        for (int r = 0; r < 8; ++r) {
            int rl  = r + half * 8;
            int row = row0 + rl;
            if (row < N) {
                float x = acc[r] + bn;
                float z = 1.0f / (1.0f + expf(-x));
                float a = Aa[rl * LDS_STRIDE + n];
                float p = As[rl * LDS_STRIDE + n];
                out[(size_t)row * D + n] = z * a + (1.0f - z) * p;
            }
        }
    }
}

// ---------------------------------------------------------------------------
extern "C" void kernel_launch(void* const* d_in, const int* in_sizes, int n_in,
                              void* d_out, int out_size, void* d_ws, size_t ws_size,
                              hipStream_t stream) {
    const float* occ3   = (const float*)d_in[0];
    const int*   idx3   = (const int*)  d_in[1];
    const float* occ5   = (const float*)d_in[2];
    const int*   idx5   = (const int*)  d_in[3];
    const float* prev   = (const float*)d_in[4];
    const float* attn_W = (const float*)d_in[5];
    const float* attn_b = (const float*)d_in[6];
    const float* gate_W = (const float*)d_in[7];
    const float* gate_b = (const float*)d_in[8];

    const int N = out_size / D;
    const long long E3 = in_sizes[1];
    const long long E5 = in_sizes[3];
    const long long E  = E3 + E5;

    float* out = (float*)d_out;

    // Workspace carve-out: keys | scores(ex) | seg_max | denom
    char* ws = (char*)d_ws;
    size_t off = 0;
    auto carve = [&](size_t bytes) {
        void* p = ws + off;
        off = (off + bytes + 255) & ~(size_t)255;
        return p;
    };
    float* keys    = (float*)carve((size_t)N * D * sizeof(float));
    float* scores  = (float*)carve((size_t)E * sizeof(float));
    float* seg_max = (float*)carve((size_t)N * sizeof(float));
    float* denom   = (float*)carve((size_t)N * sizeof(float));
    (void)ws_size; (void)n_in;

    {   // init
        size_t tot = (size_t)N * D;
        int blocks = (int)((tot + 255) / 256);
        init_kernel<<<blocks, 256, 0, stream>>>(out, seg_max, denom, N);
    }
    keys_kernel<<<(N + 15) / 16, 256, 0, stream>>>(prev, attn_W, attn_b, keys, N);
    scores_kernel<<<(int)((E + 7) / 8), 256, 0, stream>>>(occ3, occ5, idx3, idx5,
                                                          keys, scores, seg_max, E3, E);
    ex_kernel<<<(int)((E + 255) / 256), 256, 0, stream>>>(idx3, idx5, scores,
                                                          seg_max, denom, E3, E);
    agg_kernel<<<(int)((E + 7) / 8), 256, 0, stream>>>(occ3, occ5, idx3, idx5,
                                                       scores, denom, out, E3, E);
    gate_kernel<<<(N + 15) / 16, 256, 0, stream>>>(prev, gate_W, gate_b, out, N);
}